// DGRL_249108103955
// MI455X (gfx1250) — compile-verified
//
#include <hip/hip_runtime.h>
#include <hip/hip_bf16.h>
#include <math.h>

// Griffin-Lim on MI455X (gfx1250): DFTs expressed as f32 WMMA GEMMs.
// B=4, F=129, T=1000, n_fft=256, hop=32, 32 iterations.

#define GL_B     4
#define GL_F     129
#define GL_T     1000
#define GL_NFFT  256
#define GL_HOP   32
#define GL_L     ((GL_T - 1) * GL_HOP + GL_NFFT)   // 32224
#define GL_M     (GL_B * GL_T)                     // 4000 frames (M dim)
#define GL_KP    132                               // istft K (129 padded to mult of 4)
#define GL_NP    144                               // stft  N (129 padded to mult of 16)
#define GL_ITER  32

// workspace layout (float elements)
#define OFF_REX    0
#define OFF_IMX    (OFF_REX    + GL_M * GL_KP)          //  528000
#define OFF_FRAMES (OFF_IMX    + GL_M * GL_KP)          // 1056000
#define OFF_WAV    (OFF_FRAMES + GL_M * GL_NFFT)        // 2080000
#define OFF_PHASE  (OFF_WAV    + GL_B * GL_L)           // 2208896
#define OFF_CI     (OFF_PHASE  + GL_B * GL_F * GL_T)    // 2724896
#define OFF_SI     (OFF_CI     + GL_KP * GL_NFFT)
#define OFF_CS     (OFF_SI     + GL_KP * GL_NFFT)
#define OFF_SS     (OFF_CS     + GL_NFFT * GL_NP)
#define OFF_WIN    (OFF_SS     + GL_NFFT * GL_NP)

typedef __attribute__((ext_vector_type(2))) float v2f;
typedef __attribute__((ext_vector_type(8))) float v8f;

__device__ __forceinline__ v8f wmma_f32(v2f a, v2f b, v8f c) {
  // D = A(16x4 f32) * B(4x16 f32) + C(16x16 f32)
  return __builtin_amdgcn_wmma_f32_16x16x4_f32(
      /*neg_a=*/false, a, /*neg_b=*/false, b,
      /*c_mod=*/(short)0, c, /*reuse_a=*/false, /*reuse_b=*/false);
}

// ---------------------------------------------------------------- tables ----
// Ci[f][n] = wf/256 * cos(2pi f n/256); Si[f][n] = -wf/256 * sin(2pi f n/256)
// Cs[k][f] = cos(2pi f k/256);          Ss[k][f] = -sin(2pi f k/256)
// win[n]   = 0.5 - 0.5 cos(2pi n/256)
__global__ void gl_init_tables(float* Ci, float* Si, float* Cs, float* Ss, float* winT) {
  const int i = blockIdx.x * blockDim.x + threadIdx.x;
  const float TWO_PI = 6.28318530717958647692f;
  const int N_I = GL_KP * GL_NFFT;       // 33792
  const int N_S = GL_NFFT * GL_NP;       // 36864
  if (i < N_I) {
    const int f = i >> 8, n = i & 255;
    float c = 0.0f, s = 0.0f;
    if (f < GL_F) {
      const float wf = (f == 0 || f == 128) ? 1.0f : 2.0f;
      const int fn = (f * n) & 255;                  // exact mod-256 phase
      float sv, cv;
      sincosf(TWO_PI * (float)fn * (1.0f / 256.0f), &sv, &cv);
      c =  wf * cv * (1.0f / 256.0f);
      s = -wf * sv * (1.0f / 256.0f);
    }
    Ci[i] = c; Si[i] = s;
  } else if (i < N_I + N_S) {
    const int j = i - N_I;
    const int k = j / GL_NP, f = j - k * GL_NP;
    float c = 0.0f, s = 0.0f;
    if (f < GL_F) {
      const int fk = (f * k) & 255;
      float sv, cv;
      sincosf(TWO_PI * (float)fk * (1.0f / 256.0f), &sv, &cv);
      c = cv; s = -sv;
    }
    Cs[j] = c; Ss[j] = s;
  } else if (i < N_I + N_S + GL_NFFT) {
    const int n = i - (N_I + N_S);
    winT[n] = 0.5f - 0.5f * cosf(TWO_PI * (float)n * (1.0f / 256.0f));
  }
}

// ------------------------------------------------------------------ spec ----
// ReX[m][f] = mag*cos(ph), ImX[m][f] = mag*sin(ph); K padded with zeros.
__global__ void gl_spec(const float* __restrict__ mag, const float* __restrict__ ph,
                        float* __restrict__ ReX, float* __restrict__ ImX) {
  const int i = blockIdx.x * blockDim.x + threadIdx.x;
  if (i >= GL_M * GL_KP) return;
  const int m = i / GL_KP, f = i - m * GL_KP;
  float re = 0.0f, im = 0.0f;
  if (f < GL_F) {
    const int b = m / GL_T, t = m - b * GL_T;
    const int src = (b * GL_F + f) * GL_T + t;
    const float mg = mag[src];
    float sv, cv;
    sincosf(ph[src], &sv, &cv);
    re = mg * cv; im = mg * sv;
  }
  ReX[i] = re; ImX[i] = im;
}

// ------------------------------------------------------- istft WMMA GEMM ----
// frames[m][n] = win[n] * ( ReX[m][:] @ Ci[:,n] + ImX[m][:] @ Si[:,n] )
// one wave per 16x16 tile; 250 M-tiles x 16 N-tiles = 4000 tiles
__global__ void __launch_bounds__(256) gl_istft(
    const float* __restrict__ ReX, const float* __restrict__ ImX,
    const float* __restrict__ Ci,  const float* __restrict__ Si,
    const float* __restrict__ winT, float* __restrict__ frames) {
  const int wave = threadIdx.x >> 5, lane = threadIdx.x & 31;
  const int tile = blockIdx.x * 8 + wave;
  if (tile >= (GL_M / 16) * (GL_NFFT / 16)) return;     // wave-uniform
  const int mt = tile >> 4, nt = tile & 15;
  const int m0 = mt << 4, n0 = nt << 4;
  const int lrow = lane & 15, lhi = lane >> 4;

  // A: vgpr picks K lsb, lane-half picks K msb of the 4-wide K slab
  const float* ar = ReX + (m0 + lrow) * GL_KP + 2 * lhi;
  const float* ai = ImX + (m0 + lrow) * GL_KP + 2 * lhi;
  // B: mirrored layout, column n0+lrow, row k + 2*lhi (+1)
  const float* bc = Ci + (2 * lhi) * GL_NFFT + n0 + lrow;
  const float* bs = Si + (2 * lhi) * GL_NFFT + n0 + lrow;

  v8f acc = {0.f, 0.f, 0.f, 0.f, 0.f, 0.f, 0.f, 0.f};
#pragma unroll 4
  for (int k = 0; k < GL_KP; k += 4) {
    v2f a_re; a_re.x = ar[k];             a_re.y = ar[k + 1];
    v2f a_im; a_im.x = ai[k];             a_im.y = ai[k + 1];
    v2f b_c;  b_c.x  = bc[k * GL_NFFT];   b_c.y  = bc[(k + 1) * GL_NFFT];
    v2f b_s;  b_s.x  = bs[k * GL_NFFT];   b_s.y  = bs[(k + 1) * GL_NFFT];
    acc = wmma_f32(a_re, b_c, acc);
    acc = wmma_f32(a_im, b_s, acc);
  }

  const float w = winT[n0 + lrow];
  float* out = frames + n0 + lrow;
#pragma unroll
  for (int r = 0; r < 8; ++r) {
    const int m = m0 + r + (lhi << 3);    // D layout: lanes16-31 hold M=r+8
    out[m * GL_NFFT] = acc[r] * w;
  }
}

// ------------------------------------------------------------------- OLA ----
// wav[b][l] = sum_t frames[b*T+t][l-32t] / max(sum_t win[l-32t]^2, 1e-8)
__global__ void gl_ola(const float* __restrict__ frames, const float* __restrict__ winT,
                       float* __restrict__ wav) {
  const int i = blockIdx.x * blockDim.x + threadIdx.x;
  if (i >= GL_B * GL_L) return;
  const int b = i / GL_L, l = i - b * GL_L;
  int tmax = l >> 5; if (tmax > GL_T - 1) tmax = GL_T - 1;
  int d = l - (GL_NFFT - 1);
  int tmin = (d <= 0) ? 0 : ((d + GL_HOP - 1) >> 5);
  float s1 = 0.0f, s2 = 0.0f;
  for (int t = tmin; t <= tmax; ++t) {
    const int n = l - (t << 5);
    const float w = winT[n];
    s1 += frames[(b * GL_T + t) * GL_NFFT + n];
    s2 += w * w;
  }
  wav[i] = s1 / fmaxf(s2, 1e-8f);
}

// -------------------------------------------------------- stft WMMA GEMM ----
// Re[m][f] = sum_k wav[b][32t+k]*win[k]*cos(2pi f k/256); Im uses -sin.
// phase = atan2(Im, Re + 1e-6). 250 M-tiles x 9 N-tiles = 2250 tiles.
__global__ void __launch_bounds__(256) gl_stft(
    const float* __restrict__ wav, const float* __restrict__ Cs,
    const float* __restrict__ Ss,  const float* __restrict__ winT,
    float* __restrict__ phase) {
  const int wave = threadIdx.x >> 5, lane = threadIdx.x & 31;
  const int tile = blockIdx.x * 8 + wave;
  if (tile >= (GL_M / 16) * (GL_NP / 16)) return;       // wave-uniform
  const int mt = tile / 9, nt = tile - mt * 9;
  const int m0 = mt << 4, n0 = nt << 4;
  const int lrow = lane & 15, lhi = lane >> 4;

  const int m = m0 + lrow;
  const int b = m / GL_T, t = m - b * GL_T;
  const float* src = wav + b * GL_L + (t << 5) + 2 * lhi;
  const float* wt  = winT + 2 * lhi;
  const int f = n0 + lrow;
  const float* bc = Cs + (2 * lhi) * GL_NP + f;
  const float* bs = Ss + (2 * lhi) * GL_NP + f;

  v8f accR = {0.f, 0.f, 0.f, 0.f, 0.f, 0.f, 0.f, 0.f};
  v8f accI = {0.f, 0.f, 0.f, 0.f, 0.f, 0.f, 0.f, 0.f};
#pragma unroll 4
  for (int k = 0; k < GL_NFFT; k += 4) {
    v2f a;   a.x   = src[k] * wt[k];      a.y   = src[k + 1] * wt[k + 1];
    v2f b_c; b_c.x = bc[k * GL_NP];       b_c.y = bc[(k + 1) * GL_NP];
    v2f b_s; b_s.x = bs[k * GL_NP];       b_s.y = bs[(k + 1) * GL_NP];
    accR = wmma_f32(a, b_c, accR);
    accI = wmma_f32(a, b_s, accI);
  }

  if (f < GL_F) {
#pragma unroll
    for (int r = 0; r < 8; ++r) {
      const int mm = m0 + r + (lhi << 3);
      const int bb = mm / GL_T, tt = mm - bb * GL_T;
      phase[(bb * GL_F + f) * GL_T + tt] = atan2f(accI[r], accR[r] + 1e-6f);
    }
  }
}

// ------------------------------------------------------------------ crop ----
__global__ void gl_crop(const float* __restrict__ wav, float* __restrict__ out) {
  const int i = blockIdx.x * blockDim.x + threadIdx.x;
  if (i >= GL_B * GL_T) return;
  const int b = i / GL_T, t = i - b * GL_T;
  out[i] = wav[b * GL_L + ((GL_HOP - 1) / 2) + t];
}

// ---------------------------------------------------------------- launch ----
extern "C" void kernel_launch(void* const* d_in, const int* in_sizes, int n_in,
                              void* d_out, int out_size, void* d_ws, size_t ws_size,
                              hipStream_t stream) {
  const float* mag = (const float*)d_in[0];
  const float* ph0 = (const float*)d_in[1];
  float* out = (float*)d_out;
  float* ws  = (float*)d_ws;

  float* ReX    = ws + OFF_REX;
  float* ImX    = ws + OFF_IMX;
  float* frames = ws + OFF_FRAMES;
  float* wav    = ws + OFF_WAV;
  float* phase  = ws + OFF_PHASE;
  float* Ci     = ws + OFF_CI;
  float* Si     = ws + OFF_SI;
  float* Cs     = ws + OFF_CS;
  float* Ss     = ws + OFF_SS;
  float* winT   = ws + OFF_WIN;

  const int initN = GL_KP * GL_NFFT + GL_NFFT * GL_NP + GL_NFFT;
  gl_init_tables<<<(initN + 255) / 256, 256, 0, stream>>>(Ci, Si, Cs, Ss, winT);

  const int specN = GL_M * GL_KP;
  const int istftBlocks = (GL_M / 16) * (GL_NFFT / 16) / 8;            // 500
  const int olaN = GL_B * GL_L;
  const int stftBlocks = ((GL_M / 16) * (GL_NP / 16) + 7) / 8;         // 282

  for (int it = 0; it < GL_ITER; ++it) {
    const float* phsrc = (it == 0) ? ph0 : phase;
    gl_spec<<<(specN + 255) / 256, 256, 0, stream>>>(mag, phsrc, ReX, ImX);
    gl_istft<<<istftBlocks, 256, 0, stream>>>(ReX, ImX, Ci, Si, winT, frames);
    gl_ola<<<(olaN + 255) / 256, 256, 0, stream>>>(frames, winT, wav);
    if (it != GL_ITER - 1)  // last iteration's phase update is dead in the reference
      gl_stft<<<stftBlocks, 256, 0, stream>>>(wav, Cs, Ss, winT, phase);
  }

  gl_crop<<<(GL_B * GL_T + 255) / 256, 256, 0, stream>>>(wav, out);
}